// SimpleGraphSageLayer_83476984365555
// MI455X (gfx1250) — compile-verified
//
#include <hip/hip_runtime.h>
#include <hip/hip_bf16.h>
#include <stddef.h>

// ---------------------------------------------------------------------------
// SimpleGraphSageLayer for MI455X (gfx1250, wave32, WMMA)
//
//   hs = h * norm
//   c  = mean_{edges e: dst(e)=v} hs[src(e)]
//   bundle = [hs | c] @ W + b          -> 20000x256 @ 256x128 GEMM (WMMA f32)
//   bundle = bundle / max(||bundle||2, 1e-12)
//   out    = BN(bundle * norm) ; return h + out
//
// Edge phase is the roofline term (~0.7 GB, L2-resident scatter atomics).
// GEMM uses V_WMMA_F32_16X16X4_F32 (full f32 precision). The K loop is split
// into an hs half and a c_sum half so the hot loop is branch-free.
// ---------------------------------------------------------------------------

#define N_NODES 20000
#define N_EDGES 640000
#define D 128
#define NTILES (N_NODES / 16)     // 1250 node tiles of 16 rows
#define GEMM_BLOCKS 80            // blocks for node_gemm (8 waves each)

typedef float v2f __attribute__((ext_vector_type(2)));
typedef float v8f __attribute__((ext_vector_type(8)));

// ---------------------------------------------------------------------------
// Kernel 1: hs = h*norm ; zero c_sum and deg (re-zero on every graph replay)
// ---------------------------------------------------------------------------
__global__ __launch_bounds__(256) void prep_kernel(const float* __restrict__ h,
                                                   const float* __restrict__ norm,
                                                   float* __restrict__ hs,
                                                   float* __restrict__ c_sum,
                                                   float* __restrict__ deg) {
  int i = blockIdx.x * blockDim.x + threadIdx.x;   // one float4 per thread
  if (i < N_NODES * (D / 4)) {
    float nr = norm[i >> 5];                       // 32 float4 per node row
    float4 hv = reinterpret_cast<const float4*>(h)[i];
    float4 r;
    r.x = hv.x * nr; r.y = hv.y * nr; r.z = hv.z * nr; r.w = hv.w * nr;
    reinterpret_cast<float4*>(hs)[i] = r;
    float4 z; z.x = 0.f; z.y = 0.f; z.z = 0.f; z.w = 0.f;
    reinterpret_cast<float4*>(c_sum)[i] = z;
  }
  if (i < N_NODES) deg[i] = 0.0f;
}

// ---------------------------------------------------------------------------
// Kernel 2: pre-swizzle W[256x128] into per-lane B-fragment order.
// For K-step kb=4*ks, col-tile n, lane L (half=L/16):
//   frag = { W[kb+2*half+0][16n+L%16], W[kb+2*half+1][16n+L%16] }
// stored at float2 index ((ks*8)+n)*32 + L  -> one b64 load per fragment.
// ---------------------------------------------------------------------------
__global__ __launch_bounds__(256) void wswz_kernel(const float* __restrict__ W,
                                                   float* __restrict__ wsw) {
  int j = blockIdx.x * blockDim.x + threadIdx.x;
  if (j >= 64 * 8 * 32) return;
  int L  = j & 31;
  int n  = (j >> 5) & 7;
  int ks = j >> 8;
  int half = L >> 4;
  int col  = 16 * n + (L & 15);
  int row0 = 4 * ks + 2 * half;
  float2 o;
  o.x = W[row0 * D + col];
  o.y = W[(row0 + 1) * D + col];
  reinterpret_cast<float2*>(wsw)[j] = o;
}

// ---------------------------------------------------------------------------
// Kernel 3: edge scatter. One wave32 per edge; lane moves one float4,
// 4x global_atomic_add_f32 into c_sum[dst]; lane 0 counts the degree.
// c_sum (10 MB) is L2-resident -> atomics resolve on-chip.
// ---------------------------------------------------------------------------
__global__ __launch_bounds__(256) void edge_kernel(const int* __restrict__ src,
                                                   const int* __restrict__ dst,
                                                   const float* __restrict__ hs,
                                                   float* __restrict__ c_sum,
                                                   float* __restrict__ deg) {
  int wid  = (blockIdx.x * blockDim.x + threadIdx.x) >> 5;   // edge id
  int lane = threadIdx.x & 31;
  if (wid >= N_EDGES) return;
  int s = src[wid];
  int t = dst[wid];
  float4 v = reinterpret_cast<const float4*>(hs + (size_t)s * D)[lane];
  float* cp = c_sum + (size_t)t * D + lane * 4;
  unsafeAtomicAdd(cp + 0, v.x);
  unsafeAtomicAdd(cp + 1, v.y);
  unsafeAtomicAdd(cp + 2, v.z);
  unsafeAtomicAdd(cp + 3, v.w);
  if (lane == 0) unsafeAtomicAdd(deg + t, 1.0f);
}

// ---------------------------------------------------------------------------
// Kernel 4: WMMA GEMM + bias + row L2-norm + *norm + BN partial stats.
// One wave per 16-node tile. A (16x256) is formed on the fly:
//   K half 0 (cols 0..127):    hs[row][col]
//   K half 1 (cols 128..255):  c_sum[row][col-128] * invdeg[row]
// Accumulators: 8 column tiles of v_wmma_f32_16x16x4_f32, 2x32 K steps.
// C/D layout: VGPR v, lane L -> row = v + 8*(L/16), col = 16n + L%16.
// ---------------------------------------------------------------------------
__global__ __launch_bounds__(256) void node_gemm_kernel(
    const float* __restrict__ hs, const float* __restrict__ c_sum,
    const float* __restrict__ deg, const float* __restrict__ wsw,
    const float* __restrict__ bias, const float* __restrict__ norm,
    float* __restrict__ outbuf, float* __restrict__ psum,
    float* __restrict__ psq) {
  __shared__ float sstat[2 * D];   // [0..127]=sum, [128..255]=sumsq

  int tid   = threadIdx.x;
  int lane  = tid & 31;
  int wv    = tid >> 5;            // wave index in block (0..7)
  int col16 = lane & 15;
  int half  = lane >> 4;

  sstat[tid] = 0.0f;
  __syncthreads();

  float biasr[8];
#pragma unroll
  for (int n = 0; n < 8; ++n) biasr[n] = bias[16 * n + col16];

  float sumAcc[8] = {0.f, 0.f, 0.f, 0.f, 0.f, 0.f, 0.f, 0.f};
  float sqAcc[8]  = {0.f, 0.f, 0.f, 0.f, 0.f, 0.f, 0.f, 0.f};

  const float2* wp = reinterpret_cast<const float2*>(wsw);
  const int totalWaves = GEMM_BLOCKS * 8;
  for (int tile = blockIdx.x * 8 + wv; tile < NTILES; tile += totalWaves) {
    const int node0 = tile * 16;
    const int rowA  = node0 + col16;                 // A-matrix row of this lane
    const float invd = 1.0f / fmaxf(deg[rowA], 1.0f);

    v8f acc[8] = {};                                 // 16x128 f32 accumulator

    // ---- K half 0: A from hs (cols 0..127), branch-free hot loop --------
    const float* aBase0 = hs + (size_t)rowA * D + 2 * half;
#pragma unroll 4
    for (int ks = 0; ks < 32; ++ks) {
      v2f a = *reinterpret_cast<const v2f*>(aBase0 + ks * 4);
      const float2* wk = wp + (size_t)ks * 256;      // 8*32 float2 per K step
#pragma unroll
      for (int n = 0; n < 8; ++n) {
        float2 bw = wk[n * 32 + lane];
        v2f bf; bf.x = bw.x; bf.y = bw.y;
        acc[n] = __builtin_amdgcn_wmma_f32_16x16x4_f32(
            false, a, false, bf, (short)0, acc[n], false, false);
      }
    }

    // ---- K half 1: A from c_sum * invdeg (cols 128..255) ----------------
    const float* aBase1 = c_sum + (size_t)rowA * D + 2 * half;
#pragma unroll 4
    for (int ks = 0; ks < 32; ++ks) {
      float2 t = *reinterpret_cast<const float2*>(aBase1 + ks * 4);
      v2f a; a.x = t.x * invd; a.y = t.y * invd;
      const float2* wk = wp + (size_t)(ks + 32) * 256;
#pragma unroll
      for (int n = 0; n < 8; ++n) {
        float2 bw = wk[n * 32 + lane];
        v2f bf; bf.x = bw.x; bf.y = bw.y;
        acc[n] = __builtin_amdgcn_wmma_f32_16x16x4_f32(
            false, a, false, bf, (short)0, acc[n], false, false);
      }
    }

    // bias
#pragma unroll
    for (int n = 0; n < 8; ++n)
#pragma unroll
      for (int v = 0; v < 8; ++v) acc[n][v] += biasr[n];

    // per-row L2 normalize, * norm, store, accumulate BN stats
#pragma unroll
    for (int v = 0; v < 8; ++v) {
      float sq = 0.0f;
#pragma unroll
      for (int n = 0; n < 8; ++n) sq += acc[n][v] * acc[n][v];
      sq += __shfl_xor(sq, 1);
      sq += __shfl_xor(sq, 2);
      sq += __shfl_xor(sq, 4);
      sq += __shfl_xor(sq, 8);                       // stays in 16-lane half
      const int row = node0 + v + 8 * half;
      const float sc = norm[row] / fmaxf(sqrtf(sq), 1e-12f);
#pragma unroll
      for (int n = 0; n < 8; ++n) {
        float val = acc[n][v] * sc;
        outbuf[(size_t)row * D + 16 * n + col16] = val;
        sumAcc[n] += val;
        sqAcc[n]  += val * val;
      }
    }
  }

  // block-level BN partials via LDS float atomics (ds_add_f32)
#pragma unroll
  for (int n = 0; n < 8; ++n) {
    atomicAdd(&sstat[16 * n + col16], sumAcc[n]);
    atomicAdd(&sstat[D + 16 * n + col16], sqAcc[n]);
  }
  __syncthreads();
  if (tid < D)
    psum[blockIdx.x * D + tid] = sstat[tid];
  else
    psq[blockIdx.x * D + (tid - D)] = sstat[tid];
}

// ---------------------------------------------------------------------------
// Kernel 5: fold BN partials -> per-feature scale/shift
// ---------------------------------------------------------------------------
__global__ void bn_stats_kernel(const float* __restrict__ psum,
                                const float* __restrict__ psq,
                                const float* __restrict__ gamma,
                                const float* __restrict__ beta,
                                float* __restrict__ scale,
                                float* __restrict__ shift) {
  int d = threadIdx.x;
  if (d >= D) return;
  float s = 0.f, q = 0.f;
  for (int b = 0; b < GEMM_BLOCKS; ++b) {
    s += psum[b * D + d];
    q += psq[b * D + d];
  }
  const float invN = 1.0f / (float)N_NODES;
  float mu  = s * invN;
  float var = q * invN - mu * mu;
  float is  = 1.0f / sqrtf(var + 1e-5f);
  float sc  = gamma[d] * is;
  scale[d] = sc;
  shift[d] = beta[d] - mu * sc;
}

// ---------------------------------------------------------------------------
// Kernel 6: out = h + outbuf*scale + shift   (BN apply + residual)
// ---------------------------------------------------------------------------
__global__ __launch_bounds__(256) void final_kernel(const float* __restrict__ h,
                                                    const float* __restrict__ outbuf,
                                                    const float* __restrict__ scale,
                                                    const float* __restrict__ shift,
                                                    float* __restrict__ out) {
  int i = blockIdx.x * blockDim.x + threadIdx.x;   // one float4 per thread
  if (i >= N_NODES * (D / 4)) return;
  int d4 = (i & 31) * 4;
  float4 hv = reinterpret_cast<const float4*>(h)[i];
  float4 ov = reinterpret_cast<const float4*>(outbuf)[i];
  float4 r;
  r.x = hv.x + ov.x * scale[d4 + 0] + shift[d4 + 0];
  r.y = hv.y + ov.y * scale[d4 + 1] + shift[d4 + 1];
  r.z = hv.z + ov.z * scale[d4 + 2] + shift[d4 + 2];
  r.w = hv.w + ov.w * scale[d4 + 3] + shift[d4 + 3];
  reinterpret_cast<float4*>(out)[i] = r;
}

// ---------------------------------------------------------------------------
extern "C" void kernel_launch(void* const* d_in, const int* in_sizes, int n_in,
                              void* d_out, int out_size, void* d_ws, size_t ws_size,
                              hipStream_t stream) {
  const float* h     = (const float*)d_in[0];
  const float* norm  = (const float*)d_in[1];
  const int*   src   = (const int*)d_in[2];
  const int*   dst   = (const int*)d_in[3];
  const float* W     = (const float*)d_in[4];
  const float* b     = (const float*)d_in[5];
  const float* gamma = (const float*)d_in[6];
  const float* beta  = (const float*)d_in[7];
  float* out = (float*)d_out;

  char* ws = (char*)d_ws;
  const size_t NODE_BYTES = (size_t)N_NODES * D * sizeof(float);  // 10,240,000
  float* hs     = (float*)(ws);
  float* c_sum  = (float*)(ws + NODE_BYTES);
  float* outbuf = (float*)(ws + 2 * NODE_BYTES);
  float* deg    = (float*)(ws + 3 * NODE_BYTES);                  // 80 KB
  float* wsw    = (float*)(ws + 3 * NODE_BYTES + 80000);          // 128 KB
  float* psum   = (float*)(ws + 3 * NODE_BYTES + 80000 + 131072);
  float* psq    = psum + GEMM_BLOCKS * D;
  float* scale  = psq + GEMM_BLOCKS * D;
  float* shift  = scale + D;

  const int ELEM4 = N_NODES * (D / 4);           // 640000 float4s
  prep_kernel<<<(ELEM4 + 255) / 256, 256, 0, stream>>>(h, norm, hs, c_sum, deg);
  wswz_kernel<<<64, 256, 0, stream>>>(W, wsw);
  edge_kernel<<<N_EDGES / 8, 256, 0, stream>>>(src, dst, hs, c_sum, deg);
  node_gemm_kernel<<<GEMM_BLOCKS, 256, 0, stream>>>(hs, c_sum, deg, wsw, b, norm,
                                                    outbuf, psum, psq);
  bn_stats_kernel<<<1, 128, 0, stream>>>(psum, psq, gamma, beta, scale, shift);
  final_kernel<<<(ELEM4 + 255) / 256, 256, 0, stream>>>(h, outbuf, scale, shift, out);
}